// MultiScaleTriplane_Pooling_4406636446013
// MI455X (gfx1250) — compile-verified
//
#include <hip/hip_runtime.h>
#include <cmath>

typedef float v2f __attribute__((ext_vector_type(2)));
typedef float v8f __attribute__((ext_vector_type(8)));

#define AC (-0.75f)
#define TWO_PI 6.28318530717958647692f

// v_sin_f32 / v_cos_f32 compute sin(2*pi*x) / cos(2*pi*x) -> exactly the
// reference's sin(feat@B * 2pi) with NO scaling instruction needed.
__device__ __forceinline__ float sin2pi(float x) {
#if __has_builtin(__builtin_amdgcn_sinf)
    return __builtin_amdgcn_sinf(x);
#else
    return __sinf(TWO_PI * x);
#endif
}
__device__ __forceinline__ float cos2pi(float x) {
#if __has_builtin(__builtin_amdgcn_cosf)
    return __builtin_amdgcn_cosf(x);
#else
    return __cosf(TWO_PI * x);
#endif
}

__device__ __forceinline__ void cubic_w(float t, float w[4]) {
    float s = 1.0f - t;
    w[0] = AC * t * s * s;
    w[1] = (AC + 2.0f) * t * t * t - (AC + 3.0f) * t * t + 1.0f;
    w[2] = (AC + 2.0f) * s * s * s - (AC + 3.0f) * s * s + 1.0f;
    w[3] = AC * s * t * t;
}

__global__ __launch_bounds__(256) void triplane_fused_kernel(
    const float* __restrict__ coords, const int* __restrict__ iterPtr,
    const float* __restrict__ p4x, const float* __restrict__ p3x,
    const float* __restrict__ p2x, const float* __restrict__ vcx,
    const float* __restrict__ p4y, const float* __restrict__ p3y,
    const float* __restrict__ p2y, const float* __restrict__ vcy,
    const float* __restrict__ p4z, const float* __restrict__ p3z,
    const float* __restrict__ p2z, const float* __restrict__ vcz,
    const float* __restrict__ Bm, float* __restrict__ out, int N)
{
    // 8 waves/block, each wave owns a private 32x34 (padded) feat tile in LDS.
    __shared__ float sh[8 * 32 * 34];

    const int lane = threadIdx.x & 31;
    const int wid  = threadIdx.x >> 5;
    const int hi   = lane >> 4;   // half-wave select (K sub-pair / M+8)
    const int lm   = lane & 15;   // row (A) / column (B,D) within tile
    const int waveBase = blockIdx.x * 256 + wid * 32;
    const int pc = min(waveBase + lane, N - 1);   // clamp tail; stores masked later

    const int it = *iterPtr;      // uniform scalar load (graph-capture safe)

    // Preload projection matrix in WMMA-B layout (no 2*pi fold: v_sin does it).
    // B operand for k-block kb, vgpr v: B[4*kb + 2*hi + v][lm]
    float bk[16];
    #pragma unroll
    for (int k = 0; k < 8; ++k) {
        bk[2 * k + 0] = Bm[(4 * k + 2 * hi + 0) * 16 + lm];
        bk[2 * k + 1] = Bm[(4 * k + 2 * hi + 1) * 16 + lm];
    }

    const float gx = coords[3 * pc + 0];
    const float gy = coords[3 * pc + 1];
    const float gz = coords[3 * pc + 2];

    float acc[32];
    #pragma unroll
    for (int c = 0; c < 32; ++c) acc[c] = 0.0f;

    // ---- 9 bicubic plane samples: 3 scale levels x 3 orientations ----
    #pragma unroll 1
    for (int lvl = 0; lvl < 3; ++lvl) {
        if (lvl == 1 && it <= 3000) continue;
        if (lvl == 2 && it <= 4000) continue;
        const int   H   = (lvl == 0) ? 32 : ((lvl == 1) ? 64 : 128);
        const int   HW  = H * H;
        const float fac = (lvl == 0) ? 1.0f : ((lvl == 1) ? 0.7f : 0.5f);
        #pragma unroll 1
        for (int ax = 0; ax < 3; ++ax) {
            const float* pl;
            if (lvl == 0)      pl = (ax == 0) ? p4x : ((ax == 1) ? p4y : p4z);
            else if (lvl == 1) pl = (ax == 0) ? p3x : ((ax == 1) ? p3y : p3z);
            else               pl = (ax == 0) ? p2x : ((ax == 1) ? p2y : p2z);
            // x-plane:(gx,gy)  y-plane:(gy,gz)  z-plane:(gx,gz)
            const float g1 = (ax == 1) ? gy : gx;
            const float g2 = (ax == 0) ? gy : gz;

            float ixp = (g1 + 1.0f) * 0.5f * (float)(H - 1);
            float iyp = (g2 + 1.0f) * 0.5f * (float)(H - 1);
            float x0f = floorf(ixp), y0f = floorf(iyp);
            float wx[4], wy[4];
            cubic_w(ixp - x0f, wx);
            cubic_w(iyp - y0f, wy);
            int x0 = (int)x0f, y0 = (int)y0f;
            int xi[4], yi[4];
            #pragma unroll
            for (int j = 0; j < 4; ++j) {
                xi[j] = min(max(x0 - 1 + j, 0), H - 1);
                yi[j] = min(max(y0 - 1 + j, 0), H - 1);
            }
            #pragma unroll 1
            for (int i = 0; i < 4; ++i) {
                const float wyi = fac * wy[i];
                const float* r0 = pl + yi[i] * H + xi[0];
                const float* r1 = pl + yi[i] * H + xi[1];
                const float* r2 = pl + yi[i] * H + xi[2];
                const float* r3 = pl + yi[i] * H + xi[3];
                #pragma unroll
                for (int c = 0; c < 32; ++c) {
                    const int o = c * HW;
                    float s;
                    s = wx[0] * r0[o];
                    s = fmaf(wx[1], r1[o], s);
                    s = fmaf(wx[2], r2[o], s);
                    s = fmaf(wx[3], r3[o], s);
                    acc[c] = fmaf(wyi, s, acc[c]);
                }
            }
        }
    }

    // ---- 3 cubic 1-D vector samples (W==1 collapses to pure y-cubic) ----
    if (it > 5000) {
        #pragma unroll 1
        for (int ax = 0; ax < 3; ++ax) {
            const float* vp = (ax == 0) ? vcx : ((ax == 1) ? vcy : vcz);
            const float  g  = (ax == 0) ? gx  : ((ax == 1) ? gy  : gz);
            float iyv = (g + 1.0f) * 0.5f * 255.0f;
            float y0f = floorf(iyv);
            float wv[4];
            cubic_w(iyv - y0f, wv);
            int y0 = (int)y0f;
            #pragma unroll 1
            for (int i = 0; i < 4; ++i) {
                int   yv = min(max(y0 - 1 + i, 0), 255);
                float w  = 0.3f * wv[i];
                const float* q = vp + yv;
                #pragma unroll
                for (int c = 0; c < 32; ++c)
                    acc[c] = fmaf(w, q[c * 256], acc[c]);
            }
        }
    }

    // ---- Stage feat tile to LDS (stride 34: bank-conflict-free, 8B aligned) ----
    float* myrow = sh + (wid * 32 + lane) * 34;
    #pragma unroll
    for (int c = 0; c < 32; ++c) myrow[c] = acc[c];
    __syncthreads();

    // ---- proj = feat @ B via chained V_WMMA_F32_16X16X4_F32, then sin/cos ----
    const float* wbase = sh + wid * 32 * 34;
    const bool fullWave = (waveBase + 32 <= N);   // wave-uniform: scalar branch
    #pragma unroll 1
    for (int T = 0; T < 2; ++T) {
        // A operand: lane holds feat[T*16 + lm][4k + 2*hi + v]
        const float* arow = wbase + (T * 16 + lm) * 34 + 2 * hi;
        v8f cc = {0.f, 0.f, 0.f, 0.f, 0.f, 0.f, 0.f, 0.f};
        #pragma unroll
        for (int k = 0; k < 8; ++k) {
            v2f a; a.x = arow[4 * k]; a.y = arow[4 * k + 1];
            v2f b; b.x = bk[2 * k];   b.y = bk[2 * k + 1];
            cc = __builtin_amdgcn_wmma_f32_16x16x4_f32(
                     /*neg_a=*/false, a, /*neg_b=*/false, b,
                     /*c_mod=*/(short)0, cc, /*reuse_a=*/false, /*reuse_b=*/false);
        }
        // D layout: VGPR r -> M = r + 8*hi, N = lm
        const int rowBase = waveBase + T * 16 + 8 * hi;
        if (fullWave) {
            #pragma unroll
            for (int r = 0; r < 8; ++r) {
                float pr = cc[r];
                float* o = out + (rowBase + r) * 32 + lm;
                o[0]  = sin2pi(pr);
                o[16] = cos2pi(pr);
            }
        } else {
            #pragma unroll
            for (int r = 0; r < 8; ++r) {
                int row = rowBase + r;
                if (row < N) {
                    float pr = cc[r];
                    out[row * 32 + lm]      = sin2pi(pr);
                    out[row * 32 + 16 + lm] = cos2pi(pr);
                }
            }
        }
    }
}

extern "C" void kernel_launch(void* const* d_in, const int* in_sizes, int n_in,
                              void* d_out, int out_size, void* d_ws, size_t ws_size,
                              hipStream_t stream) {
    (void)n_in; (void)out_size; (void)d_ws; (void)ws_size;
    const float* coords = (const float*)d_in[0];
    const int*   iter   = (const int*)d_in[1];
    const float* p4x = (const float*)d_in[2];
    const float* p3x = (const float*)d_in[3];
    const float* p2x = (const float*)d_in[4];
    const float* vcx = (const float*)d_in[5];
    const float* p4y = (const float*)d_in[6];
    const float* p3y = (const float*)d_in[7];
    const float* p2y = (const float*)d_in[8];
    const float* vcy = (const float*)d_in[9];
    const float* p4z = (const float*)d_in[10];
    const float* p3z = (const float*)d_in[11];
    const float* p2z = (const float*)d_in[12];
    const float* vcz = (const float*)d_in[13];
    const float* Bm  = (const float*)d_in[14];
    float* out = (float*)d_out;

    const int N = in_sizes[0] / 3;
    dim3 block(256);
    dim3 grid((N + 255) / 256);
    triplane_fused_kernel<<<grid, block, 0, stream>>>(
        coords, iter,
        p4x, p3x, p2x, vcx,
        p4y, p3y, p2y, vcy,
        p4z, p3z, p2z, vcz,
        Bm, out, N);
}